// OLD_LocalGraphEmbedding_86749749444729
// MI455X (gfx1250) — compile-verified
//
#include <hip/hip_runtime.h>

typedef __attribute__((ext_vector_type(2))) float v2f;
typedef __attribute__((ext_vector_type(8))) float v8f;

#define D 64
#define TILE_M 16

// ---------------------------------------------------------------------------
// Kernel 1: zero the workspace (agg [N*64] + cnt [N]) — d_ws is poisoned.
// ---------------------------------------------------------------------------
__global__ void sage_zero_kernel(float* __restrict__ p, int n) {
    int i = blockIdx.x * blockDim.x + threadIdx.x;
    int stride = gridDim.x * blockDim.x;
    for (; i < n; i += stride) p[i] = 0.0f;
}

// ---------------------------------------------------------------------------
// Kernel 2: edge scatter. One wave32 per edge: lanes cooperatively read the
// 64-float source row (float2 per lane, coalesced 256B) and scatter-add into
// the destination row with f32 global atomics. Lane 0 bumps the degree count.
// x (25.6 MB) is L2-resident on a 192 MB L2, so this runs at L2 atomic rate.
// ---------------------------------------------------------------------------
__global__ void sage_edge_scatter_kernel(const float* __restrict__ x,
                                         const int*   __restrict__ src,
                                         const int*   __restrict__ dst,
                                         float*       __restrict__ agg,
                                         float*       __restrict__ cnt,
                                         int n_edges) {
    const int wavesPerBlock = blockDim.x >> 5;
    const int edge = blockIdx.x * wavesPerBlock + (threadIdx.x >> 5);
    if (edge >= n_edges) return;                    // wave-uniform branch
    const int lane = threadIdx.x & 31;

    const int s = src[edge];
    const int d = dst[edge];

    const float2 v = reinterpret_cast<const float2*>(x + (size_t)s * D)[lane];
    float* ar = agg + (size_t)d * D + lane * 2;
    atomicAdd(ar + 0, v.x);
    atomicAdd(ar + 1, v.y);
    if (lane == 0) atomicAdd(cnt + d, 1.0f);
}

// ---------------------------------------------------------------------------
// Kernel 3: fused mean + dual GEMM + bias + ReLU via V_WMMA_F32_16X16X4_F32.
// One wave per 16-row output tile. A-frag layout (ISA 7.12.2, 32-bit A 16x4):
//   lane l -> M = l&15, VGPR v -> K = 2*(l>>4) + v.
// B-frag (4x16, row striped across lanes): lane l -> N = n0 + (l&15),
//   VGPR v -> K = 2*(l>>4) + v.
// C/D 16x16: VGPR r: lanes 0-15 -> M=r, lanes 16-31 -> M=r+8; N = lane&15.
// ---------------------------------------------------------------------------
__global__ void sage_wmma_gemm_kernel(const float* __restrict__ x,
                                      const float* __restrict__ agg,
                                      const float* __restrict__ cnt,
                                      const float* __restrict__ Wl,
                                      const float* __restrict__ bias,
                                      const float* __restrict__ Wr,
                                      float*       __restrict__ out,
                                      int n_tiles, int n_nodes) {
    const int wavesPerBlock = blockDim.x >> 5;
    const int tile = blockIdx.x * wavesPerBlock + (threadIdx.x >> 5);
    if (tile >= n_tiles) return;                    // wave-uniform: EXEC stays all-1s
    const int lane = threadIdx.x & 31;
    const int m    = lane & 15;
    const int kh   = (lane >> 4) << 1;              // 0 or 2

    int row = tile * TILE_M + m;
    if (row >= n_nodes) row = n_nodes - 1;          // safe clamp (N%16==0 here)

    const float  c   = cnt[row];
    const float  inv = 1.0f / fmaxf(c, 1.0f);
    const float* xrow = x   + (size_t)row * D;
    const float* arow = agg + (size_t)row * D;

    v8f acc[4] = {v8f{}, v8f{}, v8f{}, v8f{}};      // 4 N-tiles of 16 cols

#pragma unroll
    for (int kk = 0; kk < D; kk += 4) {
        const int ka = kk + kh;
        v2f a_mean, a_self;
        a_mean[0] = arow[ka + 0] * inv;
        a_mean[1] = arow[ka + 1] * inv;
        a_self[0] = xrow[ka + 0];
        a_self[1] = xrow[ka + 1];

#pragma unroll
        for (int nt = 0; nt < 4; ++nt) {
            const int ncol = nt * 16 + m;
            v2f b_l, b_r;
            b_l[0] = Wl[(size_t)(ka + 0) * D + ncol];
            b_l[1] = Wl[(size_t)(ka + 1) * D + ncol];
            b_r[0] = Wr[(size_t)(ka + 0) * D + ncol];
            b_r[1] = Wr[(size_t)(ka + 1) * D + ncol];
            // mean_nbr @ W_l
            acc[nt] = __builtin_amdgcn_wmma_f32_16x16x4_f32(
                false, a_mean, false, b_l, (short)0, acc[nt], false, false);
            // x @ W_r
            acc[nt] = __builtin_amdgcn_wmma_f32_16x16x4_f32(
                false, a_self, false, b_r, (short)0, acc[nt], false, false);
        }
    }

    // Epilogue: bias + ReLU, write 16x64 tile.
    const int mbase = tile * TILE_M + ((lane >> 4) << 3);   // +8 for hi half-wave
#pragma unroll
    for (int nt = 0; nt < 4; ++nt) {
        const int   ncol = nt * 16 + m;
        const float b    = bias[ncol];
#pragma unroll
        for (int r = 0; r < 8; ++r) {
            const int mrow = mbase + r;
            if (mrow < n_nodes) {
                out[(size_t)mrow * D + ncol] = fmaxf(acc[nt][r] + b, 0.0f);
            }
        }
    }
}

// ---------------------------------------------------------------------------
// Host-side launcher. Inputs (setup_inputs order):
//   d_in[0] x        [N,64]   f32
//   d_in[1] edge_idx [2,E]    int32 (JAX x64 disabled -> int32)
//   d_in[2] W_l      [64,64]  f32
//   d_in[3] b_l      [64]     f32
//   d_in[4] W_r      [64,64]  f32
// d_out: [N,64] f32. d_ws: agg [N*64] f32 then cnt [N] f32 (~26 MB).
// ---------------------------------------------------------------------------
extern "C" void kernel_launch(void* const* d_in, const int* in_sizes, int n_in,
                              void* d_out, int out_size, void* d_ws, size_t ws_size,
                              hipStream_t stream) {
    (void)n_in; (void)out_size; (void)ws_size;

    const float* x    = (const float*)d_in[0];
    const int*   ei   = (const int*)d_in[1];
    const float* Wl   = (const float*)d_in[2];
    const float* bl   = (const float*)d_in[3];
    const float* Wr   = (const float*)d_in[4];
    float*       out  = (float*)d_out;

    const int n_nodes = in_sizes[0] / D;
    const int n_edges = in_sizes[1] / 2;
    const int* src = ei;
    const int* dst = ei + n_edges;

    float* agg = (float*)d_ws;
    float* cnt = agg + (size_t)n_nodes * D;

    // 1) zero agg + cnt
    const int n_zero = n_nodes * D + n_nodes;
    int zb = (n_zero + 255) / 256;
    if (zb > 2048) zb = 2048;
    sage_zero_kernel<<<zb, 256, 0, stream>>>(agg, n_zero);

    // 2) edge scatter: 8 waves per 256-thread block, one edge per wave
    const int epb = 256 / 32;
    const int eb  = (n_edges + epb - 1) / epb;
    sage_edge_scatter_kernel<<<eb, 256, 0, stream>>>(x, src, dst, agg, cnt, n_edges);

    // 3) fused mean + dual WMMA GEMM + bias + ReLU
    const int n_tiles = (n_nodes + TILE_M - 1) / TILE_M;
    const int tpb = 256 / 32;
    const int gb  = (n_tiles + tpb - 1) / tpb;
    sage_wmma_gemm_kernel<<<gb, 256, 0, stream>>>(x, agg, cnt, Wl, bl, Wr, out,
                                                  n_tiles, n_nodes);
}